// selfAttention_4664334484181
// MI455X (gfx1250) — compile-verified
//
#include <hip/hip_runtime.h>

typedef float v2f __attribute__((ext_vector_type(2)));
typedef float v8f __attribute__((ext_vector_type(8)));

#define S_LEN  2048
#define BATCH  8
#define DMODEL 512
#define BN     128      // keys per KV iteration
#define QSTR   516      // LDS stride for Q tile (pad: conflict-free, 8B aligned rows)
#define PSTR   132      // LDS stride for P tile
#define NWAVE  8

// D = A*B + C, fp32 WMMA 16x16x4. A: 16x4 (2 VGPR), B: 4x16 (2 VGPR), C/D: 16x16 (8 VGPR).
// A layout: lane l -> row M = l%16, component r -> K = r + 2*(l>=16)
// B layout: lane l -> col N = l%16, component r -> K = r + 2*(l>=16)
// C layout: lane l -> col N = l%16, VGPR r -> row M = r + 8*(l>=16)

__launch_bounds__(256, 2)
__global__ void flash_attn_f32_wmma(const float* __restrict__ X,  // [S, B, D]
                                    float* __restrict__ Out)      // [B, S, D]
{
    const int tid  = threadIdx.x;
    const int wid  = tid >> 5;      // wave id 0..7 (wave32)
    const int lane = tid & 31;
    const int lm   = lane & 15;
    const int hi   = lane >> 4;

    const int b     = blockIdx.x >> 7;           // S_LEN/16 = 128 query tiles / batch
    const int qbase = (blockIdx.x & 127) << 4;   // 16 queries per block

    __shared__ float Qs[16 * QSTR];
    __shared__ float Ps[16 * PSTR];
    __shared__ float redmax[NWAVE * 16];
    __shared__ float redsum[NWAVE * 16];
    __shared__ float m_sh[16];
    __shared__ float l_sh[16];
    __shared__ float a_sh[16];

    if (tid < 16) { m_sh[tid] = -1e30f; l_sh[tid] = 0.0f; }

    // ---- stage Q tile (16 x 512) into LDS, coalesced ----
    for (int idx = tid; idx < 16 * DMODEL; idx += 256) {
        const int row = idx >> 9;
        const int col = idx & (DMODEL - 1);
        Qs[row * QSTR + col] =
            X[((size_t)(qbase + row) * BATCH + b) * DMODEL + col];
    }
    __syncthreads();

    const float scale = 0.044194173824159216f;   // 1/sqrt(512)

    v8f o0 = {}, o1 = {}, o2 = {}, o3 = {};      // O accum: 16 rows x 64 cols (4 tiles)

    const float* Qrow = &Qs[lm * QSTR + 2 * hi];

    for (int kt = 0; kt < S_LEN / BN; ++kt) {
        const int keybase = kt * BN;

        // ---- S = (Q K^T)*scale : wave w owns key cols [keybase+16w, +16) ----
        const float* Krow =
            X + ((size_t)(keybase + 16 * wid + lm) * BATCH + b) * DMODEL + 2 * hi;
        v8f s = {};
        #pragma unroll 8
        for (int kc = 0; kc < DMODEL / 4; ++kc) {
            v2f a  = *(const v2f*)(Qrow + kc * 4);
            v2f bb = *(const v2f*)(Krow + kc * 4);
            s = __builtin_amdgcn_wmma_f32_16x16x4_f32(false, a, false, bb,
                                                      (short)0, s, false, false);
        }

        // ---- per-row max of this wave's 16x16 subtile (reduce across 16 lanes) ----
        float mp[8];
        #pragma unroll
        for (int r = 0; r < 8; ++r) {
            float v = s[r] * scale;
            s[r] = v;
            #pragma unroll
            for (int off = 8; off >= 1; off >>= 1)
                v = fmaxf(v, __shfl_xor(v, off, 32));
            mp[r] = v;
        }
        if (lm == 0) {
            #pragma unroll
            for (int r = 0; r < 8; ++r) redmax[wid * 16 + r + 8 * hi] = mp[r];
        }
        __syncthreads();

        // ---- online softmax state update (m, alpha) by 16 threads ----
        if (tid < 16) {
            float mt = redmax[tid];
            #pragma unroll
            for (int w = 1; w < NWAVE; ++w) mt = fmaxf(mt, redmax[w * 16 + tid]);
            const float mo = m_sh[tid];
            const float mn = fmaxf(mo, mt);
            m_sh[tid] = mn;
            a_sh[tid] = __expf(mo - mn);
        }
        __syncthreads();

        // ---- P = exp(S - m); post P to LDS; rescale O; row-sum partials ----
        float sp[8];
        #pragma unroll
        for (int r = 0; r < 8; ++r) {
            const float mrow = m_sh[r + 8 * hi];
            const float arow = a_sh[r + 8 * hi];
            const float p = __expf(s[r] - mrow);
            Ps[(r + 8 * hi) * PSTR + 16 * wid + lm] = p;
            o0[r] *= arow; o1[r] *= arow; o2[r] *= arow; o3[r] *= arow;
            float t = p;
            #pragma unroll
            for (int off = 8; off >= 1; off >>= 1)
                t += __shfl_xor(t, off, 32);
            sp[r] = t;
        }
        if (lm == 0) {
            #pragma unroll
            for (int r = 0; r < 8; ++r) redsum[wid * 16 + r + 8 * hi] = sp[r];
        }
        __syncthreads();

        if (tid < 16) {
            float su = redsum[tid];
            #pragma unroll
            for (int w = 1; w < NWAVE; ++w) su += redsum[w * 16 + tid];
            l_sh[tid] = a_sh[tid] * l_sh[tid] + su;
        }
        // (safe: next write to redsum is after two more barriers, to a_sh after one)

        // ---- O += P * V : wave w owns output cols [64w, 64w+64) ----
        const float* Vb   = X + ((size_t)keybase * BATCH + b) * DMODEL;
        const int    dcol = 64 * wid + lm;
        #pragma unroll 4
        for (int kc = 0; kc < BN / 4; ++kc) {
            v2f a = *(const v2f*)(&Ps[lm * PSTR + kc * 4 + 2 * hi]);
            const float* Vr0 = Vb + (size_t)(kc * 4 + 2 * hi) * (BATCH * DMODEL) + dcol;
            const float* Vr1 = Vr0 + (size_t)(BATCH * DMODEL);
            v2f b0; b0.x = Vr0[0];  b0.y = Vr1[0];
            o0 = __builtin_amdgcn_wmma_f32_16x16x4_f32(false, a, false, b0, (short)0, o0, false, false);
            v2f b1; b1.x = Vr0[16]; b1.y = Vr1[16];
            o1 = __builtin_amdgcn_wmma_f32_16x16x4_f32(false, a, false, b1, (short)0, o1, false, false);
            v2f b2; b2.x = Vr0[32]; b2.y = Vr1[32];
            o2 = __builtin_amdgcn_wmma_f32_16x16x4_f32(false, a, false, b2, (short)0, o2, false, false);
            v2f b3; b3.x = Vr0[48]; b3.y = Vr1[48];
            o3 = __builtin_amdgcn_wmma_f32_16x16x4_f32(false, a, false, b3, (short)0, o3, false, false);
        }
    }

    __syncthreads();   // l_sh final value visible to all waves

    // ---- epilogue: O / l, store [B, S, D] ----
    float* Ob = Out + ((size_t)b * S_LEN + qbase) * DMODEL + 64 * wid + lm;
    #pragma unroll
    for (int r = 0; r < 8; ++r) {
        const float linv = 1.0f / l_sh[r + 8 * hi];
        float* Or = Ob + (size_t)(r + 8 * hi) * DMODEL;
        Or[0]  = o0[r] * linv;
        Or[16] = o1[r] * linv;
        Or[32] = o2[r] * linv;
        Or[48] = o3[r] * linv;
    }
}

extern "C" void kernel_launch(void* const* d_in, const int* in_sizes, int n_in,
                              void* d_out, int out_size, void* d_ws, size_t ws_size,
                              hipStream_t stream) {
    (void)in_sizes; (void)n_in; (void)out_size; (void)d_ws; (void)ws_size;
    const float* X   = (const float*)d_in[0];
    float*       Out = (float*)d_out;
    dim3 grid(BATCH * (S_LEN / 16));   // 1024 blocks
    dim3 block(256);                   // 8 wave32 waves
    hipLaunchKernelGGL(flash_attn_f32_wmma, grid, block, 0, stream, X, Out);
}